// SpatialGraphConv_89404039233647
// MI455X (gfx1250) — compile-verified
//
#include <hip/hip_runtime.h>
#include <hip/hip_bf16.h>

typedef float v2f __attribute__((ext_vector_type(2)));
typedef float v8f __attribute__((ext_vector_type(8)));

#define C_IN  64
#define C_OUT 128
#define SAMP  32

// ---------------------------------------------------------------------------
// Kernel 1: per-point neighbourhood aggregation (the 281 MB bound stream).
//   aggT[c][n] = sum_s relu(pos[n,s]·W_int[c] + b_int[c]) * neigh[3+c][n][s]
// ---------------------------------------------------------------------------
__global__ void k_aggregate(const float* __restrict__ xyz,    // (N, 3)
                            const float* __restrict__ neigh,  // (67, N, S)
                            const float* __restrict__ W_int,  // (64, 3)
                            const float* __restrict__ b_int,  // (64)
                            float* __restrict__ aggT,         // (64, N)
                            int N) {
    const int n = blockIdx.x * blockDim.x + threadIdx.x;
    if (n >= N) return;

    const float cx = xyz[n * 3 + 0];
    const float cy = xyz[n * 3 + 1];
    const float cz = xyz[n * 3 + 2];

    const size_t chanStride = (size_t)N * SAMP;
    const size_t ptOff      = (size_t)n * SAMP;

    float px[SAMP], py[SAMP], pz[SAMP];
    {
        const float4* bx = (const float4*)(neigh + 0 * chanStride + ptOff);
        const float4* by = (const float4*)(neigh + 1 * chanStride + ptOff);
        const float4* bz = (const float4*)(neigh + 2 * chanStride + ptOff);
#pragma unroll
        for (int q = 0; q < SAMP / 4; ++q) {
            float4 vx = bx[q], vy = by[q], vz = bz[q];
            px[q * 4 + 0] = vx.x - cx; px[q * 4 + 1] = vx.y - cx;
            px[q * 4 + 2] = vx.z - cx; px[q * 4 + 3] = vx.w - cx;
            py[q * 4 + 0] = vy.x - cy; py[q * 4 + 1] = vy.y - cy;
            py[q * 4 + 2] = vy.z - cy; py[q * 4 + 3] = vy.w - cy;
            pz[q * 4 + 0] = vz.x - cz; pz[q * 4 + 1] = vz.y - cz;
            pz[q * 4 + 2] = vz.z - cz; pz[q * 4 + 3] = vz.w - cz;
        }
    }

    for (int c = 0; c < C_IN; ++c) {
        const float w0 = W_int[c * 3 + 0];
        const float w1 = W_int[c * 3 + 1];
        const float w2 = W_int[c * 3 + 2];
        const float bb = b_int[c];
        const float4* fb = (const float4*)(neigh + (size_t)(3 + c) * chanStride + ptOff);
        if (c + 1 < C_IN)
            __builtin_prefetch(neigh + (size_t)(4 + c) * chanStride + ptOff, 0, 0);
        float acc = 0.f;
#pragma unroll
        for (int q = 0; q < SAMP / 4; ++q) {
            float4 f = fb[q];
            const int s = q * 4;
            float h;
            h = fmaxf(0.f, fmaf(w0, px[s + 0], fmaf(w1, py[s + 0], fmaf(w2, pz[s + 0], bb))));
            acc = fmaf(h, f.x, acc);
            h = fmaxf(0.f, fmaf(w0, px[s + 1], fmaf(w1, py[s + 1], fmaf(w2, pz[s + 1], bb))));
            acc = fmaf(h, f.y, acc);
            h = fmaxf(0.f, fmaf(w0, px[s + 2], fmaf(w1, py[s + 2], fmaf(w2, pz[s + 2], bb))));
            acc = fmaf(h, f.z, acc);
            h = fmaxf(0.f, fmaf(w0, px[s + 3], fmaf(w1, py[s + 3], fmaf(w2, pz[s + 3], bb))));
            acc = fmaf(h, f.w, acc);
        }
        aggT[(size_t)c * N + n] = acc;
    }
}

// ---------------------------------------------------------------------------
// Kernel 2 (specialized, NPTS compile-time): OUT(128 x NPTS) =
//   W(128 x 64) @ X(64 x NPTS) + bias, f32 WMMA.
// One 16x64 output strip per wave = 4 independent accumulators; each A
// fragment feeds 4 WMMAs. Compile-time NPTS folds every X row stride into
// the 24-bit immediate offset of global_load_b32 (kb*NPTS*4 <= 7.9MB < 2^23),
// so the K-loop body is loads + v_wmma only.
// Fragment layouts (ISA 7.12.2):
//   A 16x4: lanes 0-15 rows M with K=(kb,kb+1); lanes 16-31 K=(kb+2,kb+3)
//   B 4x16: lane half selects K pair, lane%16 selects column
//   C 16x16: VGPR v -> row m0+v (lanes 0-15) / m0+8+v (lanes 16-31)
// ---------------------------------------------------------------------------
template <int NPTS>
__global__ void k_gemm_wmma(const float* __restrict__ W,     // (128, 64) row-major
                            const float* __restrict__ X,     // (64, NPTS)
                            const float* __restrict__ bias,  // (128)
                            float* __restrict__ OUT) {       // (128, NPTS)
    const int lane = threadIdx.x & 31;
    const int wave = threadIdx.x >> 5;
    const int tile = blockIdx.x * (blockDim.x >> 5) + wave;
    const int mt = tile & 7;        // 8 tiles over M=128
    const int ng = tile >> 3;       // group of 64 columns
    const int m0 = mt * 16;
    const int n0 = ng * 64;
    const int half = lane >> 4;
    const int l    = lane & 15;

    v8f acc[4];
#pragma unroll
    for (int t = 0; t < 4; ++t)
#pragma unroll
        for (int v = 0; v < 8; ++v)
            acc[t][v] = bias[m0 + v + half * 8];

    const float* wp = W + (size_t)(m0 + l) * C_IN + 2 * half;
    const float* xp = X + (size_t)(2 * half) * NPTS + n0 + l;  // single base, imm offsets

#pragma unroll
    for (int kk = 0; kk < C_IN / 4; ++kk) {
        const int kb = kk * 4;
        v2f a = *(const v2f*)(wp + kb);
#pragma unroll
        for (int t = 0; t < 4; ++t) {
            v2f b;
            b.x = xp[(size_t)kb * NPTS + t * 16];
            b.y = xp[(size_t)(kb + 1) * NPTS + t * 16];
            acc[t] = __builtin_amdgcn_wmma_f32_16x16x4_f32(false, a, false, b,
                                                           (short)0, acc[t], false, false);
        }
    }

#pragma unroll
    for (int t = 0; t < 4; ++t)
#pragma unroll
        for (int v = 0; v < 8; ++v)
            OUT[(size_t)(m0 + v + half * 8) * NPTS + n0 + t * 16 + l] = acc[t][v];
}

// Runtime-Npts fallback (1 tile/wave) — only used if N != 32768.
__global__ void k_gemm_wmma_dyn(const float* __restrict__ W, const float* __restrict__ X,
                                const float* __restrict__ bias, float* __restrict__ OUT,
                                int Npts) {
    const int lane = threadIdx.x & 31;
    const int wave = threadIdx.x >> 5;
    const int tile = blockIdx.x * (blockDim.x >> 5) + wave;
    const int m0 = (tile & 7) * 16;
    const int n0 = (tile >> 3) * 16;
    const int half = lane >> 4;
    const int l    = lane & 15;

    v8f c;
#pragma unroll
    for (int v = 0; v < 8; ++v)
        c[v] = bias[m0 + v + half * 8];

    const float* wp = W + (size_t)(m0 + l) * C_IN + 2 * half;
#pragma unroll
    for (int kk = 0; kk < C_IN / 4; ++kk) {
        const int kb = kk * 4;
        v2f a = *(const v2f*)(wp + kb);
        const size_t xo = (size_t)(kb + 2 * half) * Npts + n0 + l;
        v2f b;
        b.x = X[xo];
        b.y = X[xo + Npts];
        c = __builtin_amdgcn_wmma_f32_16x16x4_f32(false, a, false, b,
                                                  (short)0, c, false, false);
    }
#pragma unroll
    for (int v = 0; v < 8; ++v)
        OUT[(size_t)(m0 + v + half * 8) * Npts + n0 + l] = c[v];
}

// ---------------------------------------------------------------------------
// Kernel 3: per-channel BN folding: scale = g*rsqrt(var+eps), shift = b-mean*scale
// (biased variance, matches jnp.var). Block b<128 -> y channel, else r channel.
// ---------------------------------------------------------------------------
__global__ void k_stats(const float* __restrict__ yp, const float* __restrict__ rp,
                        const float* __restrict__ g_y, const float* __restrict__ be_y,
                        const float* __restrict__ g_r, const float* __restrict__ be_r,
                        float* __restrict__ scale, float* __restrict__ shift, int Npts) {
    const int cidx = blockIdx.x;  // 0..255
    const bool isY = cidx < C_OUT;
    const float* src = isY ? (yp + (size_t)cidx * Npts)
                           : (rp + (size_t)(cidx - C_OUT) * Npts);
    float s = 0.f, s2 = 0.f;
    for (int i = threadIdx.x; i < Npts; i += blockDim.x) {
        float v = src[i];
        s += v;
        s2 = fmaf(v, v, s2);
    }
    __shared__ float sh[256], sh2[256];
    sh[threadIdx.x] = s;
    sh2[threadIdx.x] = s2;
    __syncthreads();
    for (int off = 128; off > 0; off >>= 1) {
        if ((int)threadIdx.x < off) {
            sh[threadIdx.x]  += sh[threadIdx.x + off];
            sh2[threadIdx.x] += sh2[threadIdx.x + off];
        }
        __syncthreads();
    }
    if (threadIdx.x == 0) {
        const float inv  = 1.0f / (float)Npts;
        const float mean = sh[0] * inv;
        const float var  = fmaxf(0.f, sh2[0] * inv - mean * mean);
        const float g  = isY ? g_y[cidx]  : g_r[cidx - C_OUT];
        const float be = isY ? be_y[cidx] : be_r[cidx - C_OUT];
        const float sc = g * rsqrtf(var + 1e-5f);
        scale[cidx] = sc;
        shift[cidx] = be - mean * sc;
    }
}

// ---------------------------------------------------------------------------
// Kernel 4: out[c][n] = relu(y*sy+ty) + relu(r*sr+tr), (128, N) layout.
// ---------------------------------------------------------------------------
__global__ void k_apply(const float* __restrict__ yp, const float* __restrict__ rp,
                        const float* __restrict__ scale, const float* __restrict__ shift,
                        float* __restrict__ out, int Npts) {
    const size_t i = (size_t)blockIdx.x * blockDim.x + threadIdx.x;
    if (i >= (size_t)C_OUT * Npts) return;
    const int c = (int)(i / (size_t)Npts);
    const float yv = fmaxf(0.f, fmaf(yp[i], scale[c], shift[c]));
    const float rv = fmaxf(0.f, fmaf(rp[i], scale[C_OUT + c], shift[C_OUT + c]));
    out[i] = yv + rv;
}

extern "C" void kernel_launch(void* const* d_in, const int* in_sizes, int n_in,
                              void* d_out, int out_size, void* d_ws, size_t ws_size,
                              hipStream_t stream) {
    const float* xyz       = (const float*)d_in[0];   // (N, 3)
    const float* feats     = (const float*)d_in[1];   // (64, N)
    const float* neigh     = (const float*)d_in[2];   // (67, N, 32)
    const float* W_int     = (const float*)d_in[3];
    const float* b_int     = (const float*)d_in[4];
    const float* W_feat    = (const float*)d_in[5];
    const float* b_feat    = (const float*)d_in[6];
    const float* g_feat    = (const float*)d_in[7];
    const float* beta_feat = (const float*)d_in[8];
    const float* W_res     = (const float*)d_in[9];
    const float* b_res     = (const float*)d_in[10];
    const float* g_res     = (const float*)d_in[11];
    const float* beta_res  = (const float*)d_in[12];

    const int N = in_sizes[0] / 3;  // 32768

    float* ws    = (float*)d_ws;
    float* aggT  = ws;                          // 64 * N
    float* y_pre = aggT  + (size_t)C_IN  * N;   // 128 * N
    float* r_pre = y_pre + (size_t)C_OUT * N;   // 128 * N
    float* scale = r_pre + (size_t)C_OUT * N;   // 256
    float* shift = scale + 2 * C_OUT;           // 256

    // 1) aggregate neighbourhood -> aggT (64, N)
    k_aggregate<<<(N + 255) / 256, 256, 0, stream>>>(xyz, neigh, W_int, b_int, aggT, N);

    // 2) WMMA GEMMs: feature path and residual path
    if (N == 32768) {
        const int waves  = 8 * (32768 / 64);   // 16x64 strip per wave
        const int blocks = waves / 8;          // 8 waves (256 thr) per block
        k_gemm_wmma<32768><<<blocks, 256, 0, stream>>>(W_feat, aggT,  b_feat, y_pre);
        k_gemm_wmma<32768><<<blocks, 256, 0, stream>>>(W_res,  feats, b_res,  r_pre);
    } else {
        const int blocks = (8 * (N / 16)) / 8;
        k_gemm_wmma_dyn<<<blocks, 256, 0, stream>>>(W_feat, aggT,  b_feat, y_pre, N);
        k_gemm_wmma_dyn<<<blocks, 256, 0, stream>>>(W_res,  feats, b_res,  r_pre, N);
    }

    // 3) BN statistics -> folded scale/shift
    k_stats<<<2 * C_OUT, 256, 0, stream>>>(y_pre, r_pre, g_feat, beta_feat,
                                           g_res, beta_res, scale, shift, N);

    // 4) apply BN + ReLU on both paths and sum
    const size_t total = (size_t)C_OUT * N;
    k_apply<<<(unsigned)((total + 255) / 256), 256, 0, stream>>>(y_pre, r_pre, scale, shift,
                                                                 (float*)d_out, N);
}